// GIoULoss_16406775071069
// MI455X (gfx1250) — compile-verified
//
#include <hip/hip_runtime.h>

// GIoU loss (mean) over 2^23 box pairs — HBM-bound streaming reduction.
// Roofline: 268 MB read / ~0.2 GFLOP -> bandwidth bound (~11.5 us floor at
// 23.3 TB/s). Design: coalesced float4 (global_load_b128) loads, 4x-unrolled
// grid-stride with grouped loads for MLP, WGP-scope streaming prefetch,
// deterministic 2-stage tree reduction. Cross-lane (wave32) reduction uses
// V_WMMA_F32_16X16X4_F32 with B = ones to form rowsums + one xor-16 shuffle.

typedef float v2f __attribute__((ext_vector_type(2)));
typedef float v8f __attribute__((ext_vector_type(8)));

// Reduce one float per lane across a wave32. Result replicated in all lanes.
__device__ __forceinline__ float wave_reduce_sum(float s) {
#if __has_builtin(__builtin_amdgcn_wmma_f32_16x16x4_f32)
    // A (16x4): lane L<16 supplies A[L][0]=s ; lane L>=16 supplies A[L-16][2]=s
    // (ISA 32-bit A 16x4 layout), other K slots 0. B = ones(4x16).
    // D[m][n] = s[m] + s[m+16], so lane L's 8 accumulator regs hold rowsums
    // m=0..7 (L<16) or m=8..15 (L>=16); 7 adds + xor-16 shuffle finish 32->1.
    v2f a; a[0] = s;    a[1] = 0.0f;
    v2f b; b[0] = 1.0f; b[1] = 1.0f;
    v8f c = {};
    c = __builtin_amdgcn_wmma_f32_16x16x4_f32(
            /*neg_a=*/false, a, /*neg_b=*/false, b,
            /*c_mod=*/(short)0, c, /*reuse_a=*/false, /*reuse_b=*/false);
    float t = ((c[0] + c[1]) + (c[2] + c[3])) + ((c[4] + c[5]) + (c[6] + c[7]));
    t += __shfl_xor(t, 16, 32);   // combine the two half-sums
    return t;
#else
    for (int off = 16; off > 0; off >>= 1) s += __shfl_xor(s, off, 32);
    return s;
#endif
}

__device__ __forceinline__ float giou_loss_elem(float4 p, float4 t) {
    // intersection
    float ix1 = fmaxf(p.x, t.x), iy1 = fmaxf(p.y, t.y);
    float ix2 = fminf(p.z, t.z), iy2 = fminf(p.w, t.w);
    float iw  = fmaxf(ix2 - ix1, 0.0f), ih = fmaxf(iy2 - iy1, 0.0f);
    float inter = iw * ih;
    // areas
    float pa = (p.z - p.x) * (p.w - p.y);
    float ta = (t.z - t.x) * (t.w - t.y);
    float uni = pa + ta - inter;
    float iou = inter / uni;
    // smallest enclosing box
    float cx1 = fminf(p.x, t.x), cy1 = fminf(p.y, t.y);
    float cx2 = fmaxf(p.z, t.z), cy2 = fmaxf(p.w, t.w);
    float cw  = fmaxf(cx2 - cx1, 0.0f), ch = fmaxf(cy2 - cy1, 0.0f);
    float convex = cw * ch;
    // faithful to reference: union_area uses iou*pred_area
    float union_area = pa + ta - iou * pa;
    float giou = iou - (convex - union_area) / convex;
    return 1.0f - giou;
}

// Stage 1: each block writes one partial sum (scaled) to dst[blockIdx.x].
// scale = 1.0f normally; the no-workspace fallback launches 1 block with
// dst = d_out and scale = 1/N so the single partial IS the mean.
__global__ void __launch_bounds__(256)
giou_partial_kernel(const float4* __restrict__ pred,
                    const float4* __restrict__ tgt,
                    float* __restrict__ dst,
                    int n, float scale) {
    const int tid    = blockIdx.x * blockDim.x + threadIdx.x;
    const int stride = gridDim.x * blockDim.x;

    float acc = 0.0f;
    int i = tid;

    // Bulk: 4x unrolled, loads grouped before compute for memory-level
    // parallelism (8 independent global_load_b128 in flight per iteration).
    // For the standard launch (n % (4*stride) == 0) this covers everything
    // with zero divergence; the tail loop below runs 0 iterations.
    for (; i + 3 * stride < n; i += 4 * stride) {
        // Streaming prefetch one unrolled tile ahead, WGP scope (locality 3
        // -> pull into all cache levels). Branchless in-bounds clamp.
        int pf = i + 4 * stride;
        pf = (pf + 3 * stride < n) ? pf : i;
        __builtin_prefetch((const void*)(pred + pf), 0, 3);
        __builtin_prefetch((const void*)(tgt  + pf), 0, 3);

        float4 p0 = pred[i];
        float4 p1 = pred[i +     stride];
        float4 p2 = pred[i + 2 * stride];
        float4 p3 = pred[i + 3 * stride];
        float4 t0 = tgt [i];
        float4 t1 = tgt [i +     stride];
        float4 t2 = tgt [i + 2 * stride];
        float4 t3 = tgt [i + 3 * stride];

        acc += giou_loss_elem(p0, t0);
        acc += giou_loss_elem(p1, t1);
        acc += giou_loss_elem(p2, t2);
        acc += giou_loss_elem(p3, t3);
    }
    // Tail (only for non-power-of-two fallback launches).
    for (; i < n; i += stride)
        acc += giou_loss_elem(pred[i], tgt[i]);

    // wave32 reduction via WMMA (EXEC all-1s here: loops have reconverged)
    float wsum = wave_reduce_sum(acc);

    __shared__ float sm[8];                         // 256 threads = 8 waves
    const int lane = threadIdx.x & 31;
    const int wave = threadIdx.x >> 5;
    if (lane == 0) sm[wave] = wsum;
    __syncthreads();
    if (threadIdx.x == 0) {
        float b = 0.0f;
        const int nw = blockDim.x >> 5;
        for (int w = 0; w < nw; ++w) b += sm[w];    // fixed order -> deterministic
        dst[blockIdx.x] = b * scale;
    }
}

// Stage 2: single block folds P partials, writes mean.
__global__ void __launch_bounds__(256)
giou_final_kernel(const float* __restrict__ partial, int p,
                  float* __restrict__ out, double inv_n) {
    float acc = 0.0f;
    for (int i = threadIdx.x; i < p; i += blockDim.x) acc += partial[i];

    float wsum = wave_reduce_sum(acc);

    __shared__ float sm[8];
    const int lane = threadIdx.x & 31;
    const int wave = threadIdx.x >> 5;
    if (lane == 0) sm[wave] = wsum;
    __syncthreads();
    if (threadIdx.x == 0) {
        float b = 0.0f;
        const int nw = blockDim.x >> 5;
        for (int w = 0; w < nw; ++w) b += sm[w];
        out[0] = (float)((double)b * inv_n);
    }
}

extern "C" void kernel_launch(void* const* d_in, const int* in_sizes, int n_in,
                              void* d_out, int out_size, void* d_ws, size_t ws_size,
                              hipStream_t stream) {
    const float4* pred = (const float4*)d_in[0];
    const float4* tgt  = (const float4*)d_in[1];
    const int n = in_sizes[0] / 4;          // boxes are [N,4] float32

    const int BLOCK = 256;                  // 8 wave32s per block
    int nblocks = 2048;                     // 2^19 threads: 16 elems/thread,
                                            // exactly 4 unrolled iterations
    size_t pcap = ws_size / sizeof(float);
    if ((size_t)nblocks > pcap) nblocks = (int)pcap;

    if (nblocks >= 1) {
        giou_partial_kernel<<<nblocks, BLOCK, 0, stream>>>(
            pred, tgt, (float*)d_ws, n, 1.0f);
        giou_final_kernel<<<1, BLOCK, 0, stream>>>(
            (const float*)d_ws, nblocks, (float*)d_out, 1.0 / (double)n);
    } else {
        // Pathological no-workspace fallback: one block computes the mean directly.
        giou_partial_kernel<<<1, BLOCK, 0, stream>>>(
            pred, tgt, (float*)d_out, n, (float)(1.0 / (double)n));
    }
}